// MiltiRegionRNN_64321430224938
// MI455X (gfx1250) — compile-verified
//
#include <hip/hip_runtime.h>
#include <math.h>

// MI455X / gfx1250, wave32. All matrix math via V_WMMA_F32_16X16X4_F32
// (fp32 in/out matches the fp32 reference; workload is serial-latency/L2
// bound, not matrix-peak bound, so full fp32 precision is free).

typedef __attribute__((ext_vector_type(2))) float v2f;
typedef __attribute__((ext_vector_type(4))) float v4f;
typedef __attribute__((ext_vector_type(8))) float v8f;
typedef int v4i __attribute__((vector_size(16)));  // matches async-lds builtin proto
typedef __attribute__((address_space(3))) v4i* lds_v4i_ptr;

#define T_LEN 128
#define B_SZ  32
#define I_SZ  128
#define H_SZ  128
#define R_SZ  100
#define O_SZ  64
#define BH    (B_SZ * H_SZ)      // 4096
#define RBH   (R_SZ * BH)        // 409600 floats = 1.6 MB
#define KTOT  384                // I + H + H
#define NK    (KTOT / 4)         // 96 wmma k-steps
#define WPACK_PER_R (8 * NK * 64)  // ntile * kstep * (16 lanes * 4 k) = 49152

static __device__ __forceinline__ v8f wmma_f32(v2f a, v2f b, v8f c) {
  // 8-arg pattern: (neg_a, A, neg_b, B, c_mod, C, reuse_a, reuse_b)
  return __builtin_amdgcn_wmma_f32_16x16x4_f32(false, a, false, b, (short)0, c,
                                               false, false);
}

__global__ void init_zero_kernel(float* __restrict__ p, int n) {
  for (int i = blockIdx.x * blockDim.x + threadIdx.x; i < n;
       i += gridDim.x * blockDim.x)
    p[i] = 0.0f;
}

// One-time weight repack into WMMA B-fragment order:
//   Wpack[r][nt][k0][n][kk] = Wcat[r][k0*4+kk][nt*16+n]
// where Wcat stacks W_ih (rows 0..127), W_hh (128..255), W_rhh (256..383).
// In the step kernel a wave then fetches its whole B fragment as one
// coalesced global_load_b64 from a contiguous 256B block.
__global__ __launch_bounds__(512) void pack_w_kernel(
    const float* __restrict__ W_ih, const float* __restrict__ W_hh,
    const float* __restrict__ W_rhh, float* __restrict__ Wpack) {
  const int r = blockIdx.x;  // 100 blocks
  for (int e = threadIdx.x; e < WPACK_PER_R; e += 512) {
    const int kk = e & 3;
    const int n  = (e >> 2) & 15;
    const int k0 = (e >> 6) % NK;
    const int nt = (e >> 6) / NK;
    const int krow = k0 * 4 + kk;        // 0..383
    const int p = krow >> 7;             // phase
    const int kloc = krow & 127;
    const float* Wsrc = (p == 0) ? W_ih : (p == 1) ? W_hh : W_rhh;
    Wpack[(size_t)r * WPACK_PER_R + e] =
        Wsrc[(size_t)r * I_SZ * H_SZ + kloc * H_SZ + nt * 16 + n];
  }
}

// Fused kernel, launched once per timestep:
//   blocks [0, msg_wgs):          msg[j,bh] = sum_i C[i,j] * Hprev[i,bh]
//   blocks [msg_wgs, msg_wgs+8):  out[b,o] = b_out[o] + sum_{r,h} Hprev[r,b,h]*W_out[r*H+h,o]
// (the out part consumes Hprev = H_{t-1}, i.e. produces out row t-1)
__global__ __launch_bounds__(512) void msg_out_kernel(
    const float* __restrict__ Hprev, float* __restrict__ msg,
    const float* __restrict__ C, const float* __restrict__ W_out,
    const float* __restrict__ b_out, float* __restrict__ outdst, int msg_wgs) {
  const int tid  = threadIdx.x;
  const int lane = tid & 31;
  const int wv   = tid >> 5;        // wave 0..15
  const int half = lane >> 4;       // 0: K={k0,k0+1}, 1: K={k0+2,k0+3}
  const int lm   = lane & 15;
  __shared__ float red[16 * 256];   // split-K reduction for the out part

  if ((int)blockIdx.x < msg_wgs) {
    // ---- msg = C^T x Hprev : M = j (pad 100->112), N = bh (4096), K = i (100)
    const int gw = blockIdx.x * 16 + wv;      // global wave 0..255
#pragma unroll 1
    for (int s = 0; s < 7; ++s) {
      const int tile = gw + 256 * s;          // 0..1791 = 7 jt-tiles x 256 bh-tiles
      const int jt = tile >> 8;               // 0..6
      const int bt = tile & 255;
      const int j  = jt * 16 + lm;
      const int jc = (j < R_SZ) ? j : (R_SZ - 1);  // clamp to stay in-bounds
      const int bh = bt * 16 + lm;
      v8f acc = {};
      for (int k0 = 0; k0 < R_SZ; k0 += 4) {  // exactly 25 wmma
        const int ko = k0 + 2 * half;
        v2f a, b;
        const float a0 = C[ko * R_SZ + jc];
        const float a1 = C[(ko + 1) * R_SZ + jc];
        a.x = (j < R_SZ) ? a0 : 0.0f;         // lane-select, no EXEC divergence
        a.y = (j < R_SZ) ? a1 : 0.0f;
        b.x = Hprev[ko * BH + bh];
        b.y = Hprev[(ko + 1) * BH + bh];
        acc = wmma_f32(a, b, acc);
      }
#pragma unroll
      for (int v = 0; v < 8; ++v) {
        const int jr = jt * 16 + v + 8 * half;  // C/D layout: M = vgpr + 8*(lane>=16)
        if (jr < R_SZ) msg[jr * BH + bt * 16 + lm] = acc[v];
      }
    }
  } else if (outdst != nullptr) {
    // ---- out tile: M = b (32 -> 2 tiles), N = o (64 -> 4 tiles), K = R*H = 12800
    const int tileid = (int)blockIdx.x - msg_wgs;  // 0..7
    const int mt = tileid >> 2;                    // 0..1
    const int nt = tileid & 3;                     // 0..3
    // split-K over 16 waves: wave w owns regions [7w, 7w+7) clipped to 100
    const int r0   = wv * 7;
    const int rend = (r0 + 7 < R_SZ) ? (r0 + 7) : R_SZ;
    v8f acc = {};
    for (int r = r0; r < rend; ++r) {
      const float* Ar = Hprev + r * BH + (mt * 16 + lm) * H_SZ;  // row b, cols h
      const float* Br = W_out + r * H_SZ * O_SZ + nt * 16 + lm;  // row h, col o
      for (int k0 = 0; k0 < H_SZ; k0 += 4) {  // 32 wmma per region
        const int ko = k0 + 2 * half;
        v2f a, b;
        a.x = Ar[ko];
        a.y = Ar[ko + 1];
        b.x = Br[ko * O_SZ];
        b.y = Br[(ko + 1) * O_SZ];
        acc = wmma_f32(a, b, acc);
      }
    }
#pragma unroll
    for (int v = 0; v < 8; ++v) red[wv * 256 + lane * 8 + v] = acc[v];
    __syncthreads();
    if (tid < 256) {
      float s = 0.0f;
#pragma unroll
      for (int w = 0; w < 16; ++w) s += red[w * 256 + tid];
      const int l = tid >> 3, v = tid & 7;
      const int M = v + 8 * (l >> 4);
      const int N = l & 15;
      const int b = mt * 16 + M;
      const int o = nt * 16 + N;
      outdst[b * O_SZ + o] = s + b_out[o];
    }
  }
}

// One workgroup per region r:
//   Hnext[r] = tanh( [x_t | Hprev[r] | msg[r]] (32x384) @ Wcat[r] (384x128) + bias[r] )
// A staged in LDS (pitch 388 floats: 16B-aligned rows, stride mod 64 banks = 4
// -> conflict-free b64 A-fragment reads). 16 waves, one 16x16 tile each,
// 96 wmma along K; B comes from the packed layout as one b64 per wmma.
#define LDA 388
__global__ __launch_bounds__(512) void step_kernel(
    const float* __restrict__ x_t, const float* __restrict__ Hprev,
    const float* __restrict__ msg, const float* __restrict__ Wpack,
    const float* __restrict__ bias, float* __restrict__ Hnext) {
  const int r    = blockIdx.x;
  const int tid  = threadIdx.x;
  const int lane = tid & 31;
  const int wv   = tid >> 5;
  const int half = lane >> 4;
  const int lm   = lane & 15;
  __shared__ __align__(16) float Asm[B_SZ * LDA];

  // ---- stage A = [x_t | Hprev[r] | msg[r]] into LDS, 16B granules
  {
    const v4f* xs = (const v4f*)x_t;
    const v4f* hs = (const v4f*)(Hprev + r * BH);
    const v4f* ms = (const v4f*)(msg + r * BH);
#if __has_builtin(__builtin_amdgcn_global_load_async_to_lds_b128)
    for (int idx = tid; idx < 1024; idx += 512) {
      const int b = idx >> 5, q = idx & 31;  // row b, 16B-chunk q
      __builtin_amdgcn_global_load_async_to_lds_b128(
          (v4i*)(xs + idx), (lds_v4i_ptr)(Asm + b * LDA + q * 4), 0, 0);
      __builtin_amdgcn_global_load_async_to_lds_b128(
          (v4i*)(hs + idx), (lds_v4i_ptr)(Asm + b * LDA + 128 + q * 4), 0, 0);
      __builtin_amdgcn_global_load_async_to_lds_b128(
          (v4i*)(ms + idx), (lds_v4i_ptr)(Asm + b * LDA + 256 + q * 4), 0, 0);
    }
#if __has_builtin(__builtin_amdgcn_s_wait_asynccnt)
    __builtin_amdgcn_s_wait_asynccnt(0);
#else
    asm volatile("s_wait_asynccnt 0x0" ::: "memory");
#endif
#else
    for (int idx = tid; idx < 1024; idx += 512) {
      const int b = idx >> 5, q = idx & 31;
      *(v4f*)(Asm + b * LDA + q * 4)       = xs[idx];
      *(v4f*)(Asm + b * LDA + 128 + q * 4) = hs[idx];
      *(v4f*)(Asm + b * LDA + 256 + q * 4) = ms[idx];
    }
#endif
  }
  __syncthreads();

  const int mt = wv >> 3;  // 0..1  (batch tile)
  const int nt = wv & 7;   // 0..7  (hidden tile)
  const float* Arow = Asm + (mt * 16 + lm) * LDA + 2 * half;
  const float* Wp = Wpack + ((size_t)r * 8 + nt) * (NK * 64) + lm * 4 + 2 * half;
  v8f acc = {};
  for (int k0 = 0; k0 < NK; ++k0) {  // 96 wmma, 1 ds_b64 + 1 global_b64 each
    const v2f a = *(const v2f*)(Arow + k0 * 4);
    const v2f b = *(const v2f*)(Wp + k0 * 64);
    acc = wmma_f32(a, b, acc);
  }

  const int n = nt * 16 + lm;
  const float bv = bias[r * H_SZ + n];
#pragma unroll
  for (int v = 0; v < 8; ++v) {
    const int m = mt * 16 + v + 8 * half;
    Hnext[r * BH + m * H_SZ + n] = tanhf(acc[v] + bv);
  }
}

extern "C" void kernel_launch(void* const* d_in, const int* in_sizes, int n_in,
                              void* d_out, int out_size, void* d_ws,
                              size_t ws_size, hipStream_t stream) {
  (void)in_sizes; (void)n_in; (void)out_size; (void)ws_size;
  const float* x     = (const float*)d_in[0];  // [T,B,I]
  const float* C     = (const float*)d_in[1];  // [R,R]
  const float* W_ih  = (const float*)d_in[2];  // [R,I,H]
  const float* W_hh  = (const float*)d_in[3];  // [R,H,H]
  const float* W_rhh = (const float*)d_in[4];  // [R,H,H]
  const float* bias  = (const float*)d_in[5];  // [R,H]
  const float* W_out = (const float*)d_in[6];  // [R*H,O]
  const float* b_out = (const float*)d_in[7];  // [O]
  float* out = (float*)d_out;                  // [T,B,O]
  float* ws  = (float*)d_ws;

  float* buf0  = ws;            // H state ping      (1.6 MB)
  float* buf1  = ws + RBH;      // H state pong      (1.6 MB)
  float* msg   = ws + 2 * RBH;  // connectome buffer (1.6 MB)
  float* Wpack = ws + 3 * RBH;  // packed weights    (19.7 MB)
  // total workspace: ~24.6 MB

  init_zero_kernel<<<256, 256, 0, stream>>>(buf0, RBH);
  pack_w_kernel<<<R_SZ, 512, 0, stream>>>(W_ih, W_hh, W_rhh, Wpack);

  for (int t = 0; t < T_LEN; ++t) {
    float* Hp = (t & 1) ? buf1 : buf0;  // holds H_{t-1}
    float* Hn = (t & 1) ? buf0 : buf1;
    float* outdst = (t == 0) ? nullptr : out + (size_t)(t - 1) * B_SZ * O_SZ;
    const int grid = (t == 0) ? 16 : 24;  // 16 msg WGs (+8 out WGs when t>0)
    msg_out_kernel<<<grid, 512, 0, stream>>>(Hp, msg, C, W_out, b_out, outdst,
                                             16);
    step_kernel<<<R_SZ, 512, 0, stream>>>(x + (size_t)t * B_SZ * I_SZ, Hp, msg,
                                          Wpack, bias, Hn);
  }
  // final output row t = T-1 (H_{T-1} lives in buf0 since T is even)
  msg_out_kernel<<<8, 512, 0, stream>>>(buf0, msg, C, W_out, b_out,
                                        out + (size_t)(T_LEN - 1) * B_SZ * O_SZ,
                                        0);
}